// PUBOsolver_19842748907975
// MI455X (gfx1250) — compile-verified
//
#include <hip/hip_runtime.h>
#include <hip/hip_bf16.h>

typedef __attribute__((ext_vector_type(2))) float v2f;
typedef __attribute__((ext_vector_type(8))) float v8f;

// ---------------------------------------------------------------------------
// 1) deg[i] = 1.0  (self-loop weight)
__global__ __launch_bounds__(256) void k_deg_init(float* __restrict__ deg, int n) {
  int i = blockIdx.x * blockDim.x + threadIdx.x;
  if (i < n) deg[i] = 1.0f;
}

// 2) deg[col[e]] += ew[e]
__global__ __launch_bounds__(256) void k_deg_acc(const long long* __restrict__ ei,
                                                 const float* __restrict__ ea,
                                                 float* __restrict__ deg, int E) {
  int e = blockIdx.x * blockDim.x + threadIdx.x;
  if (e < E) {
    int c = (int)ei[(size_t)E + e];
    atomicAdd(&deg[c], ea[e]);
  }
}

// 3) dinv[i] = deg>0 ? rsqrt(deg) : 0   (in place)
__global__ __launch_bounds__(256) void k_dinv(float* __restrict__ deg, int n) {
  int i = blockIdx.x * blockDim.x + threadIdx.x;
  if (i < n) {
    float d = deg[i];
    deg[i] = d > 0.0f ? rsqrtf(d) : 0.0f;
  }
}

// 4) hW = x @ W1 via V_WMMA_F32_16X16X4_F32; fused agg1 = dinv^2 * hW.
//    One wave = one 16x16 output tile; 8 waves/block cover all 128 cols of a
//    16-row slab (shared x rows hit L0/L2).
__global__ __launch_bounds__(256) void k_gemm1(const float* __restrict__ x,
                                               const float* __restrict__ W1,
                                               const float* __restrict__ dinv,
                                               float* __restrict__ hW,
                                               float* __restrict__ agg1,
                                               int nrows) {
  const int lane = threadIdx.x & 31;
  const int waveId = threadIdx.x >> 5;       // 0..7 -> 16-col tile
  const int m0 = blockIdx.x * 16;
  const int n0 = waveId * 16;
  const int l15 = lane & 15;
  const int hi = lane >> 4;                  // 0 or 1 (lane half)

  int mr = m0 + l15;
  if (mr >= nrows) mr = nrows - 1;           // clamp: EXEC must stay all-1s for WMMA
  const float* __restrict__ xrow = x + (size_t)mr * 256;
  const int kSel = hi * 2;                   // lanes 16-31 take K+2,K+3

  // Prefetch the back half of this lane's x row (global_prefetch_b8) while
  // the first WMMAs chew on the front half. x is the only real HBM stream.
  __builtin_prefetch(xrow + 128, 0, 3);
  __builtin_prefetch(xrow + 192, 0, 3);

  v8f c = {};
#pragma unroll 8
  for (int k0 = 0; k0 < 256; k0 += 4) {
    v2f a, b;
    // A 16x4: lane holds row m, (VGPR0,VGPR1) = x[m][k0+kSel], x[m][k0+kSel+1]
    a.x = xrow[k0 + kSel];
    a.y = xrow[k0 + kSel + 1];
    // B 4x16: lane holds col n, (VGPR0,VGPR1) = W1[k0+kSel][n], W1[k0+kSel+1][n]
    const float* __restrict__ wp = W1 + (size_t)(k0 + kSel) * 128 + n0 + l15;
    b.x = wp[0];
    b.y = wp[128];
    c = __builtin_amdgcn_wmma_f32_16x16x4_f32(false, a, false, b,
                                              (short)0, c, false, false);
  }

  // C/D 16x16 f32: VGPR v, lanes 0-15 -> M=v, lanes 16-31 -> M=v+8, N=lane&15
  const int nIdx = n0 + l15;
  const int mBase = m0 + (hi << 3);
  if (m0 + 16 <= nrows) {
    // wave-uniform fast path: no per-lane exec games (always taken for N%16==0)
#pragma unroll
    for (int v = 0; v < 8; ++v) {
      int m = mBase + v;
      float val = c[v];
      size_t off = (size_t)m * 128 + nIdx;
      float di = dinv[m];                    // uniform across each 16-lane half
      hW[off] = val;
      agg1[off] = di * di * val;
    }
  } else {
#pragma unroll
    for (int v = 0; v < 8; ++v) {
      int m = mBase + v;
      if (m < nrows) {
        float val = c[v];
        size_t off = (size_t)m * 128 + nIdx;
        float di = dinv[m];
        hW[off] = val;
        agg1[off] = di * di * val;
      }
    }
  }
}

// 5) layer-1 edge scatter: one wave per edge, float4 per lane (128 feats / 32 lanes)
__global__ __launch_bounds__(256) void k_scatter1(const long long* __restrict__ ei,
                                                  const float* __restrict__ ea,
                                                  const float* __restrict__ dinv,
                                                  const float* __restrict__ hW,
                                                  float* __restrict__ agg1, int E) {
  int e = blockIdx.x * 8 + (threadIdx.x >> 5);
  int lane = threadIdx.x & 31;
  if (e >= E) return;
  int r = (int)ei[e];
  int cN = (int)ei[(size_t)E + e];
  float coef = dinv[r] * ea[e] * dinv[cN];
  const float4 v = ((const float4*)(hW + (size_t)r * 128))[lane];
  float* dst = agg1 + (size_t)cN * 128 + lane * 4;
  atomicAdd(dst + 0, coef * v.x);
  atomicAdd(dst + 1, coef * v.y);
  atomicAdd(dst + 2, coef * v.z);
  atomicAdd(dst + 3, coef * v.w);
}

// 6) z[i] = sum_f relu(agg1[i][f]+b1[f]) * W2[f]; fused agg2 = dinv^2 * z.
//    One wave per node; wave32 shuffle reduction.
__global__ __launch_bounds__(256) void k_matvec(const float* __restrict__ agg1,
                                                const float* __restrict__ b1,
                                                const float* __restrict__ W2,
                                                const float* __restrict__ dinv,
                                                float* __restrict__ z,
                                                float* __restrict__ agg2, int n) {
  int i = blockIdx.x * 8 + (threadIdx.x >> 5);
  int lane = threadIdx.x & 31;
  if (i >= n) return;
  float4 a = ((const float4*)(agg1 + (size_t)i * 128))[lane];
  float4 bb = ((const float4*)b1)[lane];
  float4 w = ((const float4*)W2)[lane];
  float s = fmaxf(a.x + bb.x, 0.0f) * w.x
          + fmaxf(a.y + bb.y, 0.0f) * w.y
          + fmaxf(a.z + bb.z, 0.0f) * w.z
          + fmaxf(a.w + bb.w, 0.0f) * w.w;
#pragma unroll
  for (int off = 16; off > 0; off >>= 1)
    s += __shfl_down(s, off, 32);
  if (lane == 0) {
    z[i] = s;
    float di = dinv[i];
    agg2[i] = di * di * s;
  }
}

// 7) layer-2 scalar edge scatter
__global__ __launch_bounds__(256) void k_scatter2(const long long* __restrict__ ei,
                                                  const float* __restrict__ ea,
                                                  const float* __restrict__ dinv,
                                                  const float* __restrict__ z,
                                                  float* __restrict__ agg2, int E) {
  int e = blockIdx.x * blockDim.x + threadIdx.x;
  if (e >= E) return;
  int r = (int)ei[e];
  int cN = (int)ei[(size_t)E + e];
  float coef = dinv[r] * ea[e] * dinv[cN];
  atomicAdd(&agg2[cN], coef * z[r]);
}

// 8) out[i] = sigmoid(agg2[i] + b2[0])
__global__ __launch_bounds__(256) void k_sigmoid(const float* __restrict__ agg2,
                                                 const float* __restrict__ b2,
                                                 float* __restrict__ out, int n) {
  int i = blockIdx.x * blockDim.x + threadIdx.x;
  if (i < n) {
    float v = agg2[i] + b2[0];
    out[i] = 1.0f / (1.0f + expf(-v));
  }
}

// ---------------------------------------------------------------------------
extern "C" void kernel_launch(void* const* d_in, const int* in_sizes, int n_in,
                              void* d_out, int out_size, void* d_ws, size_t ws_size,
                              hipStream_t stream) {
  const long long* ei = (const long long*)d_in[0];  // [2,E] int64
  const float* ea = (const float*)d_in[1];          // [E]
  const float* x  = (const float*)d_in[2];          // [N,256]
  const float* W1 = (const float*)d_in[3];          // [256,128]
  const float* b1 = (const float*)d_in[4];          // [128]
  const float* W2 = (const float*)d_in[5];          // [128,1]
  const float* b2 = (const float*)d_in[6];          // [1]

  const int E = in_sizes[1];
  const int N = in_sizes[2] / 256;

  // workspace carve-out (all f32): dinv[N] | hW[N*128] | agg1[N*128] | z[N] | agg2[N]
  size_t npad = ((size_t)N + 255) & ~(size_t)255;
  float* dinv = (float*)d_ws;
  float* hW   = dinv + npad;
  float* agg1 = hW + (size_t)N * 128;
  float* zbuf = agg1 + (size_t)N * 128;
  float* agg2 = zbuf + npad;
  float* out  = (float*)d_out;

  int bN  = (N + 255) / 256;
  int bE  = (E + 255) / 256;

  k_deg_init<<<bN, 256, 0, stream>>>(dinv, N);
  k_deg_acc<<<bE, 256, 0, stream>>>(ei, ea, dinv, E);
  k_dinv<<<bN, 256, 0, stream>>>(dinv, N);

  int gemmBlocks = (N + 15) / 16;
  k_gemm1<<<gemmBlocks, 256, 0, stream>>>(x, W1, dinv, hW, agg1, N);

  int s1Blocks = (E + 7) / 8;  // 8 waves (edges) per 256-thread block
  k_scatter1<<<s1Blocks, 256, 0, stream>>>(ei, ea, dinv, hW, agg1, E);

  int mvBlocks = (N + 7) / 8;
  k_matvec<<<mvBlocks, 256, 0, stream>>>(agg1, b1, W2, dinv, zbuf, agg2, N);

  k_scatter2<<<bE, 256, 0, stream>>>(ei, ea, dinv, zbuf, agg2, E);
  k_sigmoid<<<bN, 256, 0, stream>>>(agg2, b2, out, N);
}